// Tooth_Assembler_58815282151851
// MI455X (gfx1250) — compile-verified
//
#include <hip/hip_runtime.h>

typedef __attribute__((ext_vector_type(2))) float v2f;
typedef __attribute__((ext_vector_type(8))) float v8f;

// One block per (b,t) pair. Thread 0 builds the augmented [R|T] 16x4 matrix
// (rows 3..15 zero) in LDS; every wave keeps it register-resident as the WMMA
// A-fragment and streams 16 points per v_wmma_f32_16x16x4_f32.
// D = [R|T] * [x;y;z;1] gives rotate+translate in one matrix op.
__global__ __launch_bounds__(256) void se3_transform_wmma(
    const float* __restrict__ X, const float* __restrict__ dof,
    float* __restrict__ out, int np)
{
    __shared__ float sA[64];               // 16 rows x 4 (K) of the A matrix
    const int tid = threadIdx.x;
    const int bt  = blockIdx.x;

    if (tid < 64) sA[tid] = 0.0f;
    __syncthreads();

    if (tid == 0) {
        const float* d = dof + (size_t)bt * 6;
        float vx = d[0], vy = d[1], vz = d[2];
        float wx = d[3], wy = d[4], wz = d[5];
        float t2 = wx*wx + wy*wy + wz*wz;
        float th = sqrtf(t2);
        bool  sm = th < 1e-4f;
        float st  = sm ? 1.0f : th;
        float st2 = sm ? 1.0f : t2;
        float st3 = sm ? 1.0f : t2 * th;
        float sth = sinf(th), cth = cosf(th);
        float Af = sm ? 1.0f - t2 * (1.0f/6.0f)        : sth / st;
        float Bf = sm ? 0.5f - t2 * (1.0f/24.0f)       : (1.0f - cth) / st2;
        float Cf = sm ? (1.0f/6.0f) - t2*(1.0f/120.0f) : (th - sth) / st3;
        float xy = wx*wy, xz = wx*wz, yz = wy*wz;
        // R = I + A*K + B*K^2, with K^2 = w w^T - t2*I
        float R00 = 1.0f + Bf*(wx*wx - t2);
        float R01 = -Af*wz + Bf*xy;
        float R02 =  Af*wy + Bf*xz;
        float R10 =  Af*wz + Bf*xy;
        float R11 = 1.0f + Bf*(wy*wy - t2);
        float R12 = -Af*wx + Bf*yz;
        float R20 = -Af*wy + Bf*xz;
        float R21 =  Af*wx + Bf*yz;
        float R22 = 1.0f + Bf*(wz*wz - t2);
        // V = I + B*K + C*K^2 ; T = V v
        float V00 = 1.0f + Cf*(wx*wx - t2);
        float V01 = -Bf*wz + Cf*xy;
        float V02 =  Bf*wy + Cf*xz;
        float V10 =  Bf*wz + Cf*xy;
        float V11 = 1.0f + Cf*(wy*wy - t2);
        float V12 = -Bf*wx + Cf*yz;
        float V20 = -Bf*wy + Cf*xz;
        float V21 =  Bf*wx + Cf*yz;
        float V22 = 1.0f + Cf*(wz*wz - t2);
        float T0 = V00*vx + V01*vy + V02*vz;
        float T1 = V10*vx + V11*vy + V12*vz;
        float T2 = V20*vx + V21*vy + V22*vz;
        sA[0]=R00; sA[1]=R01; sA[2] =R02; sA[3] =T0;
        sA[4]=R10; sA[5]=R11; sA[6] =R12; sA[7] =T1;
        sA[8]=R20; sA[9]=R21; sA[10]=R22; sA[11]=T2;
    }
    __syncthreads();

    const int  lane = tid & 31;
    const int  wave = tid >> 5;          // 0..7
    const int  r    = lane & 15;
    const bool hi   = lane >= 16;

    // A-fragment (16x4 f32, 2 VGPRs): lane r holds K0/K1; lane r+16 holds K2/K3
    v2f afrag;
    afrag[0] = sA[r*4 + (hi ? 2 : 0)];
    afrag[1] = sA[r*4 + (hi ? 3 : 1)];

    const size_t slab = (size_t)bt * (size_t)np * 3;
    const float* __restrict__ xin  = X   + slab;
    float*       __restrict__ xout = out + slab;

    const int ntiles = np >> 4;              // 16 points per tile (48 floats)
    const int off0   = hi ? r*3 + 2 : r*3;   // lo: x_n, hi: z_n
    const int off1   = r*3 + 1;              // y_n (discarded on hi lanes)
    const int step   = 8 * 48;               // 8 waves advance together

    // Strength-reduced per-lane pointers; advance by 384 floats per trip.
    const float* p = xin  + (size_t)wave * 48;
    float*       q = xout + (size_t)wave * 48 + r * 3;

    #pragma unroll 2
    for (int tile = wave; tile < ntiles; tile += 8) {
        // B-fragment (4x16): v0 = K0|K2 rows, v1 = K1|K3 rows, column = point
        float b0  = p[off0];
        float b1y = p[off1];
        v2f bfrag;
        bfrag[0] = b0;
        bfrag[1] = hi ? 1.0f : b1y;          // K3 row = homogeneous 1

        v8f c = {};
        v8f dres = __builtin_amdgcn_wmma_f32_16x16x4_f32(
            /*neg_a=*/false, afrag, /*neg_b=*/false, bfrag,
            /*c_mod=*/(short)0, c, /*reuse_a=*/false, /*reuse_b=*/false);

        if (!hi) {                            // rows 0..2 of D, lanes 0..15
            q[0] = dres[0];                   // x'  (merges to global_store_b96)
            q[1] = dres[1];                   // y'
            q[2] = dres[2];                   // z'
        }
        p += step;
        q += step;
    }
}

extern "C" void kernel_launch(void* const* d_in, const int* in_sizes, int n_in,
                              void* d_out, int out_size, void* d_ws, size_t ws_size,
                              hipStream_t stream)
{
    const float* X   = (const float*)d_in[0];
    const float* dof = (const float*)d_in[1];
    float* out = (float*)d_out;

    const int n_bt = in_sizes[1] / 6;            // B*NT = 1024
    const int np   = in_sizes[0] / (n_bt * 3);   // 8192

    se3_transform_wmma<<<n_bt, 256, 0, stream>>>(X, dof, out, np);
}